// EKF_29489245454687
// MI455X (gfx1250) — compile-verified
//
#include <hip/hip_runtime.h>
#include <hip/hip_bf16.h>
#include <math.h>

// EKF scan for MI455X (gfx1250), single persistent wave32.
// Per step: Psi·P·Psi^T done as two dependent V_WMMA_F32_16X16X4_F32 ops.
// Layout tricks:
//   * A-layout(X) == B-layout(X^T) element-for-element on CDNA5, so one v2f
//     register pair serves Psi as B (=Psi^T) in WMMA#1 and as A in WMMA#2.
//   * P is symmetric -> its A-layout is built from 9 wave-replicated scalars
//     with no cross-lane traffic.
//   * C-layout -> B-layout conversion of the intermediate is a half-wave
//     swap (lane ^ 16) -> __shfl_xor, no generic bpermute needed.
//   * 3x3 result broadcast uses v_readlane into SGPRs so the scalar EKF
//     update runs on the CDNA5 SALU float pipeline.

typedef __attribute__((ext_vector_type(2))) float v2f;
typedef __attribute__((ext_vector_type(8))) float v8f;

#define DT_  (1.0f / 262.0f)
#define ETA_ 1e-6f

__device__ __forceinline__ float softplus_f(float x) { return log1pf(__expf(x)); }

// constant-lane broadcast via v_readlane_b32 (result is wave-uniform / SGPR)
__device__ __forceinline__ float bcast(float v, int srclane) {
    return __int_as_float(__builtin_amdgcn_readlane(__float_as_int(v), srclane));
}

__global__ __launch_bounds__(32)
void ekf_wmma_kernel(const float* __restrict__ obs,
                     const float* __restrict__ garch,
                     const float* __restrict__ carma,
                     const float* __restrict__ w0v,
                     const float* __restrict__ P0,
                     const float* __restrict__ b0p,
                     const float* __restrict__ b1p,
                     const float* __restrict__ a1p,
                     const float* __restrict__ kappap,
                     const float* __restrict__ thetap,
                     const float* __restrict__ xip,
                     const float* __restrict__ rhop,
                     float* __restrict__ out, int T)
{
    const int  lane = (int)(threadIdx.x & 31u);
    const int  lo   = lane & 15;
    const bool hi   = lane >= 16;

    // loop-invariant transformed parameters
    const float b0    = b0p[0], b1 = b1p[0], a1 = a1p[0];
    const float theta = thetap[0];
    const float kap   = softplus_f(kappap[0]);
    const float xi_   = softplus_f(xip[0]);
    const float rho_  = tanhf(rhop[0]);
    const float xi2   = xi_ * xi_;
    const float f1    = b0 * DT_;          // phi_row[1]
    const float f2    = b1 * DT_;          // phi_row[2]
    const float onem_a1dt = 1.0f - a1 * DT_;

    // state: w (3) and symmetric P (9), wave-uniform scalars
    float w0 = w0v[0], w1 = w0v[1], w2 = w0v[2];
    float p00 = P0[0], p01 = P0[1], p02 = P0[2];
    float p10 = P0[3], p11 = P0[4], p12 = P0[5];
    float p20 = P0[6], p21 = P0[7], p22 = P0[8];

    const v8f zero8 = {};

    for (int t = 0; t < T; ++t) {
        if ((t & 15) == 0) {   // stream inputs ahead (global_prefetch_b8)
            __builtin_prefetch(obs   + t + 128, 0, 1);
            __builtin_prefetch(garch + t + 128, 0, 1);
            __builtin_prefetch(carma + 2 * t + 256, 0, 1);
        }

        const float ob = obs[t];
        const float g0 = garch[t];
        const float c0 = carma[2 * t];
        const float c1 = carma[2 * t + 1];

        // ---- scalar prelude ----
        float th   = softplus_f(theta + g0);
        float term = kap * th * fminf(__expf(-w0), 1.0f);
        term = (term != term) ? 0.0f : term;     // NaN guard
        const float psi00 = 1.0f - term;
        const float sigma = __expf(0.5f * w0);
        const float cross = sigma * xi_ * rho_;

        // w_pred = w + psi_vec * DT
        const float wp0 = w0 + (term + 0.5f * xi2 - kap) * DT_;
        const float wp1 = w1 + (w2 + c0) * DT_;
        const float wp2 = w2 + (-a1 * w2 + c1) * DT_;

        // ---- Psi in combined A/B layout (A-layout(Psi) == B-layout(Psi^T)) ----
        v2f psiAB;
        psiAB.x = hi ? (lo == 1 ? DT_ : (lo == 2 ? onem_a1dt : 0.0f))   // k=2 col
                     : (lo == 0 ? psi00 : 0.0f);                        // k=0 col
        psiAB.y = hi ? 0.0f : (lo == 1 ? 1.0f : 0.0f);                  // k=3 / k=1

        // ---- P in A-layout (symmetric -> built from uniform scalars) ----
        v2f aP;
        aP.x = hi ? (lo == 0 ? p02 : lo == 1 ? p12 : lo == 2 ? p22 : 0.0f)
                  : (lo == 0 ? p00 : lo == 1 ? p10 : lo == 2 ? p20 : 0.0f);
        aP.y = hi ? 0.0f
                  : (lo == 0 ? p01 : lo == 1 ? p11 : lo == 2 ? p21 : 0.0f);

        // ---- WMMA #1: N1 = P @ Psi^T ----
        v8f n1 = __builtin_amdgcn_wmma_f32_16x16x4_f32(
            false, aP, false, psiAB, (short)0, zero8, false, false);

        // C-layout -> B-layout for N1: row 2 moves to lanes 16..31, vgpr 0.
        // That is exactly a lane^16 exchange for the lanes that consume it.
        const float n1r2 = __shfl_xor(n1[2], 16);
        v2f bN1;
        bN1.x = hi ? n1r2 : n1[0];
        bN1.y = hi ? 0.0f : n1[1];

        // ---- WMMA #2: M2 = Psi @ N1 = Psi P Psi^T ----
        v8f m2 = __builtin_amdgcn_wmma_f32_16x16x4_f32(
            false, psiAB, false, bN1, (short)0, zero8, false, false);

        // broadcast the 3x3 block into wave-uniform scalars (v_readlane)
        const float m00 = bcast(m2[0], 0), m01 = bcast(m2[0], 1), m02 = bcast(m2[0], 2);
        const float m10 = bcast(m2[1], 0), m11 = bcast(m2[1], 1), m12 = bcast(m2[1], 2);
        const float m20 = bcast(m2[2], 0), m21 = bcast(m2[2], 1), m22 = bcast(m2[2], 2);

        // P_pred = sym(M2) + R + I*ETA   (R already symmetric)
        const float q00 = m00 + xi2 * DT_ + ETA_ + ETA_;
        const float q11 = m11 + ETA_ + ETA_;
        const float q22 = m22 + sigma * sigma * DT_ + ETA_ + ETA_;
        const float q01 = 0.5f * (m01 + m10);
        const float q02 = 0.5f * (m02 + m20) + cross * DT_;
        const float q12 = 0.5f * (m12 + m21);

        // measurement update
        const float x_pred = (b0 * wp1 + b1 * wp2) * DT_;
        const float sig2p  = __expf(wp0);
        const float Qd = f1 * f1 * q11 + 2.0f * f1 * f2 * q12 + f2 * f2 * q22
                       + sig2p * DT_ + ETA_ + ETA_;
        const float s0v = f1 * q01 + f2 * q02;   // (P_pred @ phi_row)
        const float s1v = f1 * q11 + f2 * q12;
        const float s2v = f1 * q12 + f2 * q22;
        const float invQ = __builtin_amdgcn_rcpf(Qd);   // v_rcp_f32, short chain
        const float K0 = s0v * invQ, K1 = s1v * invQ, K2 = s2v * invQ;
        const float innov = ob - x_pred;

        const float wn0 = wp0 + K0 * innov;
        const float wn1 = wp1 + K1 * innov;
        const float wn2 = wp2 + K2 * innov;

        // P_post = (I - K phi^T) P_pred, then symmetrize + I*ETA
        const float a00 = q00 - K0 * s0v, a01 = q01 - K0 * s1v, a02 = q02 - K0 * s2v;
        const float a10 = q01 - K1 * s0v, a11 = q11 - K1 * s1v, a12 = q12 - K1 * s2v;
        const float a20 = q02 - K2 * s0v, a21 = q12 - K2 * s1v, a22 = q22 - K2 * s2v;

        p00 = a00 + ETA_;  p11 = a11 + ETA_;  p22 = a22 + ETA_;
        p01 = 0.5f * (a01 + a10);  p10 = p01;
        p02 = 0.5f * (a02 + a20);  p20 = p02;
        p12 = 0.5f * (a12 + a21);  p21 = p12;
        w0 = wn0; w1 = wn1; w2 = wn2;

        if (lane == 0) {
            float4 o = make_float4(x_pred, wn0, wn1, wn2);
            *(float4*)(out + 4 * t) = o;
        }
        // divergence closed before next iteration's WMMAs (EXEC all-1s again)
    }
}

extern "C" void kernel_launch(void* const* d_in, const int* in_sizes, int n_in,
                              void* d_out, int out_size, void* d_ws, size_t ws_size,
                              hipStream_t stream) {
    const float* obs   = (const float*)d_in[0];
    const float* garch = (const float*)d_in[1];
    const float* carma = (const float*)d_in[2];
    const float* w0v   = (const float*)d_in[3];
    const float* P0    = (const float*)d_in[4];
    const float* b0    = (const float*)d_in[5];
    const float* b1    = (const float*)d_in[6];
    const float* a1    = (const float*)d_in[7];
    const float* kappa = (const float*)d_in[8];
    const float* theta = (const float*)d_in[9];
    const float* xi    = (const float*)d_in[10];
    const float* rho   = (const float*)d_in[11];
    const int T = in_sizes[0];

    ekf_wmma_kernel<<<1, 32, 0, stream>>>(obs, garch, carma, w0v, P0,
                                          b0, b1, a1, kappa, theta, xi, rho,
                                          (float*)d_out, T);
}